// SAGE_26809185861708
// MI455X (gfx1250) — compile-verified
//
#include <hip/hip_runtime.h>

typedef __attribute__((ext_vector_type(2))) float v2f;
typedef __attribute__((ext_vector_type(8))) float v8f;

// ---------------------------------------------------------------------------
// Zero agg+deg region (scalar-granularity; cost negligible vs edge traffic)
// ---------------------------------------------------------------------------
__global__ void k_zero(float* __restrict__ p, long n) {
    long i = (long)blockIdx.x * blockDim.x + threadIdx.x;
    if (i < n) p[i] = 0.0f;
}

// ---------------------------------------------------------------------------
// Edge scatter: agg[dst] += x[src]  (64 floats / edge, 4 floats / thread)
//               deg[dst] += 1       (one thread per edge)
// ---------------------------------------------------------------------------
__global__ void k_scatter(const float* __restrict__ X,
                          const int*   __restrict__ src,
                          const int*   __restrict__ dst,
                          float*       __restrict__ agg,
                          float*       __restrict__ deg,
                          int n_edges) {
    int tid = blockIdx.x * blockDim.x + threadIdx.x;
    int e = tid >> 4;          // 16 chunks of float4 per edge
    int c = tid & 15;
    if (e >= n_edges) return;
    int s = src[e];
    int d = dst[e];
    const float4 v = *reinterpret_cast<const float4*>(X + (size_t)s * 64 + c * 4);
    float* p = agg + (size_t)d * 64 + c * 4;
    __hip_atomic_fetch_add(p + 0, v.x, __ATOMIC_RELAXED, __HIP_MEMORY_SCOPE_AGENT);
    __hip_atomic_fetch_add(p + 1, v.y, __ATOMIC_RELAXED, __HIP_MEMORY_SCOPE_AGENT);
    __hip_atomic_fetch_add(p + 2, v.z, __ATOMIC_RELAXED, __HIP_MEMORY_SCOPE_AGENT);
    __hip_atomic_fetch_add(p + 3, v.w, __ATOMIC_RELAXED, __HIP_MEMORY_SCOPE_AGENT);
    if (c == 0)
        __hip_atomic_fetch_add(deg + d, 1.0f, __ATOMIC_RELAXED, __HIP_MEMORY_SCOPE_AGENT);
}

// ---------------------------------------------------------------------------
// In-place mean: agg[n][:] *= 1 / max(deg[n], 1)
// ---------------------------------------------------------------------------
__global__ void k_mean(float* __restrict__ agg,
                       const float* __restrict__ deg,
                       int n_nodes) {
    int tid  = blockIdx.x * blockDim.x + threadIdx.x;
    int node = tid >> 4;
    int c    = tid & 15;
    if (node >= n_nodes) return;
    float inv = 1.0f / fmaxf(deg[node], 1.0f);
    float4* p = reinterpret_cast<float4*>(agg + (size_t)node * 64 + c * 4);
    float4 v = *p;
    v.x *= inv; v.y *= inv; v.z *= inv; v.w *= inv;
    *p = v;
}

// ---------------------------------------------------------------------------
// Fused SAGE layer GEMM via V_WMMA_F32_16X16X4_F32 (full fp32 precision):
//   OUT[16 rows x DOUT] = X_tile @ Wself + MEAN_tile @ Wneigh + bias (+ReLU)
// One wave32 per 16-row tile; K=64 -> 16 k-steps of K=4.
// Weights staged in LDS in K-pair-swizzled layout so each B fragment is one
// 8B-aligned ds_load_b64 directly into an even VGPR pair (no re-pairing movs):
//   lds[(k/2)*2*DOUT + 2*col + (k&1)] = W[k][col]
// EXEC stays all-1s inside the WMMA region (tail guard is wave-uniform).
// ---------------------------------------------------------------------------
template <int DOUT, bool RELU>
__global__ __launch_bounds__(256)
void k_sage_gemm(const float* __restrict__ X,      // [N,64]  self features
                 const float* __restrict__ MEAN,   // [N,64]  mean-aggregated
                 const float* __restrict__ Ws,     // [64,DOUT] row-major
                 const float* __restrict__ Wn,     // [64,DOUT] row-major
                 const float* __restrict__ bias,   // [DOUT]
                 float*       __restrict__ OUT,    // [N,DOUT]
                 int n_tiles) {
    constexpr int NT = DOUT / 16;
    __shared__ float lws[64 * DOUT];
    __shared__ float lwn[64 * DOUT];
    for (int i = threadIdx.x; i < 64 * DOUT; i += 256) {
        int k   = i / DOUT;
        int col = i - k * DOUT;
        int idx = (k >> 1) * (2 * DOUT) + (col << 1) + (k & 1);
        lws[idx] = Ws[i];
        lwn[idx] = Wn[i];
    }
    __syncthreads();

    const int lane = threadIdx.x & 31;
    const int wave = threadIdx.x >> 5;
    const int tile = blockIdx.x * 8 + wave;
    if (tile >= n_tiles) return;                 // wave-uniform exit

    const int row0  = tile * 16;
    const int m     = lane & 15;                 // M (A) / N (B,C) sub-index
    const int khalf = (lane >> 4) * 2;           // K offset for lane half

    // Accumulators seeded with bias (broadcast across the 8 M-rows per lane).
    v8f acc[NT];
#pragma unroll
    for (int t = 0; t < NT; ++t) {
        float bv = bias[t * 16 + m];
#pragma unroll
        for (int r = 0; r < 8; ++r) acc[t][r] = bv;
    }

    const float* xrow = X    + (size_t)(row0 + m) * 64;
    const float* mrow = MEAN + (size_t)(row0 + m) * 64;

#pragma unroll
    for (int kb = 0; kb < 16; ++kb) {
        const int k = kb * 4 + khalf;            // even -> 8B-aligned loads
        // A fragments: lane L<16 -> (M=L, K=k,k+1); L>=16 -> (M=L-16, K=k+2,k+3)
        v2f a_s = *reinterpret_cast<const v2f*>(xrow + k);
        v2f a_n = *reinterpret_cast<const v2f*>(mrow + k);
        // Swizzled-LDS base for this wave-half's K pair; column tiles are at
        // constant 128B offsets -> immediate-offset ds_load_b64.
        const int kp = (k >> 1);                 // = kb*2 + (lane>>4)
        const float* ps = lws + kp * (2 * DOUT) + (m << 1);
        const float* pn = lwn + kp * (2 * DOUT) + (m << 1);
#pragma unroll
        for (int t = 0; t < NT; ++t) {
            v2f bs = *reinterpret_cast<const v2f*>(ps + t * 32);
            acc[t] = __builtin_amdgcn_wmma_f32_16x16x4_f32(
                false, a_s, false, bs, (short)0, acc[t], false, false);
            v2f bn = *reinterpret_cast<const v2f*>(pn + t * 32);
            acc[t] = __builtin_amdgcn_wmma_f32_16x16x4_f32(
                false, a_n, false, bn, (short)0, acc[t], false, false);
        }
    }

    // C/D layout: VGPR r -> row (r) for lanes 0-15, row (r+8) for lanes 16-31.
    const int rbase = row0 + (lane >> 4) * 8;
#pragma unroll
    for (int t = 0; t < NT; ++t) {
#pragma unroll
        for (int r = 0; r < 8; ++r) {
            float v = acc[t][r];
            if (RELU) v = fmaxf(v, 0.0f);
            OUT[(size_t)(rbase + r) * DOUT + t * 16 + m] = v;
        }
    }
}

// ---------------------------------------------------------------------------
// Host-side layer driver
// ---------------------------------------------------------------------------
static inline void run_layer(const float* h_in, const float* Ws, const float* Wn,
                             const float* b, float* h_out, float* agg, float* deg,
                             const int* src, const int* dst,
                             int n_nodes, int n_edges, int dout, bool relu,
                             hipStream_t stream) {
    // 1) zero agg + deg (contiguous region)
    long nz = (long)n_nodes * 64 + n_nodes;
    k_zero<<<(int)((nz + 255) / 256), 256, 0, stream>>>(agg, nz);
    // 2) edge scatter
    long nsc = (long)n_edges * 16;
    k_scatter<<<(int)((nsc + 255) / 256), 256, 0, stream>>>(h_in, src, dst, agg, deg, n_edges);
    // 3) mean
    long nm = (long)n_nodes * 16;
    k_mean<<<(int)((nm + 255) / 256), 256, 0, stream>>>(agg, deg, n_nodes);
    // 4) fused dual-GEMM + bias (+relu)
    int n_tiles = (n_nodes + 15) / 16;
    int blocks  = (n_tiles + 7) / 8;
    if (dout == 64) {
        if (relu)
            k_sage_gemm<64, true><<<blocks, 256, 0, stream>>>(h_in, agg, Ws, Wn, b, h_out, n_tiles);
        else
            k_sage_gemm<64, false><<<blocks, 256, 0, stream>>>(h_in, agg, Ws, Wn, b, h_out, n_tiles);
    } else {
        if (relu)
            k_sage_gemm<32, true><<<blocks, 256, 0, stream>>>(h_in, agg, Ws, Wn, b, h_out, n_tiles);
        else
            k_sage_gemm<32, false><<<blocks, 256, 0, stream>>>(h_in, agg, Ws, Wn, b, h_out, n_tiles);
    }
}

extern "C" void kernel_launch(void* const* d_in, const int* in_sizes, int n_in,
                              void* d_out, int out_size, void* d_ws, size_t ws_size,
                              hipStream_t stream) {
    const float* x   = (const float*)d_in[0];
    const int*   src = (const int*)d_in[1];
    const int*   dst = (const int*)d_in[2];
    const float* Ws1 = (const float*)d_in[3];
    const float* Wn1 = (const float*)d_in[4];
    const float* b1  = (const float*)d_in[5];
    const float* Ws2 = (const float*)d_in[6];
    const float* Wn2 = (const float*)d_in[7];
    const float* b2  = (const float*)d_in[8];
    const float* Ws3 = (const float*)d_in[9];
    const float* Wn3 = (const float*)d_in[10];
    const float* b3  = (const float*)d_in[11];

    const int n_nodes = in_sizes[0] / 64;   // 100000 (divisible by 16)
    const int n_edges = in_sizes[1];        // 1600000

    // Workspace layout (floats): agg[N*64] | deg[N] | h1[N*64] | h2[N*64]
    float* agg = (float*)d_ws;
    float* deg = agg + (size_t)n_nodes * 64;
    float* h1  = deg + n_nodes;
    float* h2  = h1 + (size_t)n_nodes * 64;
    float* out = (float*)d_out;

    // Layer 1: x  -> h1   (64 -> 64)
    run_layer(x,  Ws1, Wn1, b1, h1,  agg, deg, src, dst, n_nodes, n_edges, 64, false, stream);
    // Layer 2: h1 -> h2   (64 -> 64)
    run_layer(h1, Ws2, Wn2, b2, h2,  agg, deg, src, dst, n_nodes, n_edges, 64, false, stream);
    // Layer 3: h2 -> out  (64 -> 32) + ReLU
    run_layer(h2, Ws3, Wn3, b3, out, agg, deg, src, dst, n_nodes, n_edges, 32, true, stream);
}